// RecPolicy_85169201480381
// MI455X (gfx1250) — compile-verified
//
#include <hip/hip_runtime.h>

#ifndef __has_builtin
#define __has_builtin(x) 0
#endif

#if __has_builtin(__builtin_amdgcn_global_load_async_to_lds_b32) && \
    __has_builtin(__builtin_amdgcn_global_store_async_from_lds_b32)
#define ASYNC_LDS 1
#else
#define ASYNC_LDS 0
#endif

#if ASYNC_LDS
typedef __attribute__((address_space(1))) int GintT; // global i32
typedef __attribute__((address_space(3))) int LintT; // LDS i32
#endif

__device__ __forceinline__ void wait_async0() {
#if __has_builtin(__builtin_amdgcn_s_wait_asynccnt)
  __builtin_amdgcn_s_wait_asynccnt(0);
#else
  asm volatile("s_wait_asynccnt 0" ::: "memory");
#endif
}

#if __has_builtin(__builtin_amdgcn_rcpf)
__device__ __forceinline__ float rcp_f(float x) { return __builtin_amdgcn_rcpf(x); }
#else
__device__ __forceinline__ float rcp_f(float x) { return 1.0f / x; }
#endif

// tanh: prefer the CDNA5 TRANS-unit v_tanh_f32; else overflow-safe exp form.
__device__ __forceinline__ float tanh_f(float x) {
#if __has_builtin(__builtin_amdgcn_tanhf)
  return __builtin_amdgcn_tanhf(x);
#else
  return fmaf(2.0f, rcp_f(1.0f + __expf(-2.0f * x)), -1.0f);
#endif
}
// sigmoid(x) = 0.5*tanh(x/2) + 0.5 (robust for all x)
__device__ __forceinline__ float sigm_f(float x) {
#if __has_builtin(__builtin_amdgcn_tanhf)
  return fmaf(0.5f, __builtin_amdgcn_tanhf(0.5f * x), 0.5f);
#else
  return rcp_f(1.0f + __expf(-x));
#endif
}

// PyTorch GRUCell, hidden=2, input=2. wih/whh row-major [6,2].
// brz[k] = bih[k]+bhh[k] for r/z gates (k=0..3); bin/bhn = n-gate biases.
__device__ __forceinline__ void gru_step(float x0, float x1, float& h0, float& h1,
                                         const float wih[12], const float whh[12],
                                         const float brz[4], const float bin[2],
                                         const float bhn[2]) {
  float g[6], q[6];
#pragma unroll
  for (int k = 0; k < 4; ++k) {
    g[k] = fmaf(wih[2 * k], x0, fmaf(wih[2 * k + 1], x1, brz[k]));
    q[k] = fmaf(whh[2 * k], h0, whh[2 * k + 1] * h1);
  }
#pragma unroll
  for (int k = 4; k < 6; ++k) {
    g[k] = fmaf(wih[2 * k], x0, fmaf(wih[2 * k + 1], x1, bin[k - 4]));
    q[k] = fmaf(whh[2 * k], h0, fmaf(whh[2 * k + 1], h1, bhn[k - 4]));
  }
  float r0 = sigm_f(g[0] + q[0]);
  float r1 = sigm_f(g[1] + q[1]);
  float z0 = sigm_f(g[2] + q[2]);
  float z1 = sigm_f(g[3] + q[3]);
  float n0 = tanh_f(fmaf(r0, q[4], g[4]));
  float n1 = tanh_f(fmaf(r1, q[5], g[5]));
  // (1-z)*n + z*h == n + z*(h-n)
  h0 = fmaf(z0, h0 - n0, n0);
  h1 = fmaf(z1, h1 - n1, n1);
}

#define ROWS_PB 256
#define IN_F 19
#define OUT_F 7

__global__ __launch_bounds__(256) void recpolicy_kernel(
    const float* __restrict__ x,
    const float* __restrict__ up_wih, const float* __restrict__ up_whh,
    const float* __restrict__ up_bih, const float* __restrict__ up_bhh,
    const float* __restrict__ dn_wih, const float* __restrict__ dn_whh,
    const float* __restrict__ dn_bih, const float* __restrict__ dn_bhh,
    const float* __restrict__ obs_w, const float* __restrict__ obs_b,
    const float* __restrict__ out_w, const float* __restrict__ out_b,
    float* __restrict__ out, int nrows) {
  __shared__ float lx[ROWS_PB * IN_F];   // 19456 B
  __shared__ float lo[ROWS_PB * OUT_F];  //  7168 B

  const int tid = threadIdx.x;

  // ---- uniform weight loads (scalar path), overlap with async staging ----
  float uwih[12], uwhh[12], dwih[12], dwhh[12];
  float ubrz[4], dbrz[4], ubin[2], ubhn[2], dbin[2], dbhn[2];
  float ow[14];
#pragma unroll
  for (int i = 0; i < 12; ++i) {
    uwih[i] = up_wih[i]; uwhh[i] = up_whh[i];
    dwih[i] = dn_wih[i]; dwhh[i] = dn_whh[i];
  }
#pragma unroll
  for (int i = 0; i < 4; ++i) {
    ubrz[i] = up_bih[i] + up_bhh[i];
    dbrz[i] = dn_bih[i] + dn_bhh[i];
  }
  ubin[0] = up_bih[4]; ubin[1] = up_bih[5];
  ubhn[0] = up_bhh[4]; ubhn[1] = up_bhh[5];
  dbin[0] = dn_bih[4]; dbin[1] = dn_bih[5];
  dbhn[0] = dn_bhh[4]; dbhn[1] = dn_bhh[5];
#pragma unroll
  for (int i = 0; i < 14; ++i) ow[i] = obs_w[i];
  const float ob0 = obs_b[0], ob1 = obs_b[1];
  const float vw0 = out_w[0], vw1 = out_w[1], vb = out_b[0];

  // ---- stage this block's input chunk into LDS, coalesced ----
  const int in_base = blockIdx.x * (ROWS_PB * IN_F);
  int in_rem = nrows * IN_F - in_base;
  if (in_rem > ROWS_PB * IN_F) in_rem = ROWS_PB * IN_F;
  const float* gx = x + in_base;

#if ASYNC_LDS
#pragma unroll
  for (int k = 0; k < IN_F; ++k) {
    int idx = tid + k * ROWS_PB;
    if (idx < in_rem)
      __builtin_amdgcn_global_load_async_to_lds_b32(
          (GintT*)(gx + idx), (LintT*)(&lx[idx]), 0, 0);
  }
  wait_async0();
#else
#pragma unroll
  for (int k = 0; k < IN_F; ++k) {
    int idx = tid + k * ROWS_PB;
    if (idx < in_rem) lx[idx] = gx[idx];
  }
#endif
  __syncthreads();

  // ---- per-thread row compute (stride-19 LDS reads: bank-conflict-free) ----
  const float* row = &lx[tid * IN_F];

  float h0 = 0.0f, h1 = 0.0f;
  float hu0[7], hu1[7];
#pragma unroll
  for (int t = 0; t < 7; ++t) {
    gru_step(row[5 + t], row[12 + t], h0, h1, uwih, uwhh, ubrz, ubin, ubhn);
    hu0[t] = h0; hu1[t] = h1;
  }

  // h = concat(obs, h_last) @ obs_w.T + obs_b
  float hd0 = ob0, hd1 = ob1;
#pragma unroll
  for (int i = 0; i < 5; ++i) {
    hd0 = fmaf(ow[i], row[i], hd0);
    hd1 = fmaf(ow[7 + i], row[i], hd1);
  }
  hd0 = fmaf(ow[5], h0, fmaf(ow[6], h1, hd0));
  hd1 = fmaf(ow[12], h0, fmaf(ow[13], h1, hd1));

#pragma unroll
  for (int t = 0; t < 7; ++t) {
    gru_step(hu0[t], hu1[t], hd0, hd1, dwih, dwhh, dbrz, dbin, dbhn);
    lo[tid * OUT_F + t] = fmaf(vw0, hd0, fmaf(vw1, hd1, vb));
  }
  __syncthreads();

  // ---- stream results back out, coalesced ----
  const int out_base = blockIdx.x * (ROWS_PB * OUT_F);
  int out_rem = nrows * OUT_F - out_base;
  if (out_rem > ROWS_PB * OUT_F) out_rem = ROWS_PB * OUT_F;
  float* gout = out + out_base;

#if ASYNC_LDS
#pragma unroll
  for (int k = 0; k < OUT_F; ++k) {
    int idx = tid + k * ROWS_PB;
    if (idx < out_rem)
      __builtin_amdgcn_global_store_async_from_lds_b32(
          (GintT*)(gout + idx), (LintT*)(&lo[idx]), 0, 0);
  }
  wait_async0();
#else
#pragma unroll
  for (int k = 0; k < OUT_F; ++k) {
    int idx = tid + k * ROWS_PB;
    if (idx < out_rem) gout[idx] = lo[idx];
  }
#endif
}

extern "C" void kernel_launch(void* const* d_in, const int* in_sizes, int n_in,
                              void* d_out, int out_size, void* d_ws, size_t ws_size,
                              hipStream_t stream) {
  const float* x      = (const float*)d_in[0];
  const float* up_wih = (const float*)d_in[1];
  const float* up_whh = (const float*)d_in[2];
  const float* up_bih = (const float*)d_in[3];
  const float* up_bhh = (const float*)d_in[4];
  const float* dn_wih = (const float*)d_in[5];
  const float* dn_whh = (const float*)d_in[6];
  const float* dn_bih = (const float*)d_in[7];
  const float* dn_bhh = (const float*)d_in[8];
  const float* obs_w  = (const float*)d_in[9];
  const float* obs_b  = (const float*)d_in[10];
  const float* out_w  = (const float*)d_in[11];
  const float* out_b  = (const float*)d_in[12];

  const int B = in_sizes[0] / 19;
  const int blocks = (B + ROWS_PB - 1) / ROWS_PB;

  recpolicy_kernel<<<blocks, ROWS_PB, 0, stream>>>(
      x, up_wih, up_whh, up_bih, up_bhh,
      dn_wih, dn_whh, dn_bih, dn_bhh,
      obs_w, obs_b, out_w, out_b,
      (float*)d_out, B);
}